// LIFNode_13589276524700
// MI455X (gfx1250) — compile-verified
//
#include <hip/hip_runtime.h>

// LIF recurrence over T for x[B, T, D] f32 -> spikes[B, T, D] f32.
//   mem_t = TAU*mem_{t-1} + x_t ; s_t = (mem_t > VTH) ; mem_t = s_t ? 0 : mem_t
// Memory-bound streaming scan (268 MB, ~11.5 us floor @ 23.3 TB/s).
// CDNA5 path: 8-deep GLOBAL_LOAD_ASYNC_TO_LDS_B128 pipeline (ASYNCcnt),
// ds_load_b128 consume, NT b128 stores. Fallback: direct NT b128 loads.

typedef float v4f __attribute__((ext_vector_type(4)));
// Exact pointee type of the async builtin's pointer params (from hipcc
// diagnostic): GCC-vector int4.
typedef int async_b128_t __attribute__((vector_size(16)));

#define GLOBAL_AS __attribute__((address_space(1)))
#define LDS_AS    __attribute__((address_space(3)))

namespace {
constexpr float TAU = 0.2f;
constexpr float VTH = 0.5f;
constexpr int B  = 16;
constexpr int T  = 32;
constexpr int D  = 65536;
constexpr int D4 = D / 4;            // float4 columns: 16384 (power of two)
constexpr int NTHREADS = B * D4;     // 262144 threads, one per (b, d4)
constexpr int BLOCK = 256;           // 8 waves per block (wave32)
constexpr int DEPTH = 8;             // async copies in flight per wave
}

#if __has_builtin(__builtin_amdgcn_global_load_async_to_lds_b128)
#define LIF_ASYNC 1
#else
#define LIF_ASYNC 0
#endif

template <int N>
__device__ __forceinline__ void wait_asynccnt() {
#if __has_builtin(__builtin_amdgcn_s_wait_asynccnt)
    __builtin_amdgcn_s_wait_asynccnt(N);
#else
    asm volatile("s_wait_asynccnt %0" :: "i"(N));
#endif
    asm volatile("" ::: "memory");   // keep LDS reads after the wait
}

__device__ __forceinline__ v4f lif_step(v4f& mem, const v4f xv) {
    mem = TAU * mem + xv;                    // packed v_fma per pair
    v4f s;
    s.x = (mem.x > VTH) ? 1.0f : 0.0f;
    s.y = (mem.y > VTH) ? 1.0f : 0.0f;
    s.z = (mem.z > VTH) ? 1.0f : 0.0f;
    s.w = (mem.w > VTH) ? 1.0f : 0.0f;
    // hard reset: mem * (1 - s) == (mem > VTH ? 0 : mem)
    mem.x = (mem.x > VTH) ? 0.0f : mem.x;
    mem.y = (mem.y > VTH) ? 0.0f : mem.y;
    mem.z = (mem.z > VTH) ? 0.0f : mem.z;
    mem.w = (mem.w > VTH) ? 0.0f : mem.w;
    return s;
}

#if LIF_ASYNC

__device__ __forceinline__ void async_copy_b128(const v4f* gsrc, v4f* lds_dst) {
    __builtin_amdgcn_global_load_async_to_lds_b128(
        (GLOBAL_AS async_b128_t*)gsrc,
        (LDS_AS async_b128_t*)lds_dst,
        /*offset=*/0, /*cpol=*/0);
}

__global__ __launch_bounds__(BLOCK) void lif_scan_kernel(
    const float* __restrict__ x, float* __restrict__ out)
{
    // Per-thread private 16B slots; each wave only touches its own lanes'
    // slots, so ordering is purely via ASYNCcnt (no barriers needed).
    __shared__ v4f stage[DEPTH][BLOCK];      // 32 KB per block

    const int tid = threadIdx.x;
    const int v = blockIdx.x * BLOCK + tid;
    const int b  = v >> 14;                  // v / D4
    const int d4 = v & (D4 - 1);             // v % D4

    // Per-t stride = 256 KiB; 31*256 KiB fits the signed 24-bit IOFFSET.
    const v4f* __restrict__ xi =
        reinterpret_cast<const v4f*>(x) + (size_t)b * T * D4 + d4;
    v4f* __restrict__ yo =
        reinterpret_cast<v4f*>(out) + (size_t)b * T * D4 + d4;

    // Prologue: fill the pipeline with DEPTH async global->LDS b128 copies.
#pragma unroll
    for (int t = 0; t < DEPTH; ++t) {
        async_copy_b128(&xi[(size_t)t * D4], &stage[t][tid]);
    }

    v4f mem = (v4f)(0.0f);

    // Steady state: retire copy t, consume from LDS, reissue slot for t+DEPTH.
#pragma unroll
    for (int t = 0; t < T - DEPTH; ++t) {
        wait_asynccnt<DEPTH - 1>();          // copy t has landed in LDS
        const v4f xv = stage[t % DEPTH][tid];    // ds_load_b128
        const v4f s = lif_step(mem, xv);
        __builtin_nontemporal_store(s, &yo[(size_t)t * D4]);
        asm volatile("" ::: "memory");       // slot consumed before reuse
        async_copy_b128(&xi[(size_t)(t + DEPTH) * D4], &stage[t % DEPTH][tid]);
    }

    // Tail: drain the pipeline, then consume the last DEPTH slots.
    wait_asynccnt<0>();
#pragma unroll
    for (int t = T - DEPTH; t < T; ++t) {
        const v4f xv = stage[t % DEPTH][tid];
        const v4f s = lif_step(mem, xv);
        __builtin_nontemporal_store(s, &yo[(size_t)t * D4]);
    }
}

#else  // LIF_ASYNC == 0: proven direct-load fallback (round-1 kernel)

__global__ __launch_bounds__(BLOCK) void lif_scan_kernel(
    const float* __restrict__ x, float* __restrict__ out)
{
    const int v = blockIdx.x * BLOCK + threadIdx.x;
    const int b  = v >> 14;
    const int d4 = v & (D4 - 1);

    const v4f* __restrict__ xi =
        reinterpret_cast<const v4f*>(x) + (size_t)b * T * D4 + d4;
    v4f* __restrict__ yo =
        reinterpret_cast<v4f*>(out) + (size_t)b * T * D4 + d4;

    v4f mem = (v4f)(0.0f);

#pragma unroll
    for (int t0 = 0; t0 < T; t0 += 8) {
        if (t0 + 8 < T) {
            __builtin_prefetch((const void*)&xi[(size_t)(t0 + 8) * D4], 0, 0);
        }
        v4f buf[8];
#pragma unroll
        for (int k = 0; k < 8; ++k) {
            buf[k] = __builtin_nontemporal_load(&xi[(size_t)(t0 + k) * D4]);
        }
#pragma unroll
        for (int k = 0; k < 8; ++k) {
            const v4f s = lif_step(mem, buf[k]);
            __builtin_nontemporal_store(s, &yo[(size_t)(t0 + k) * D4]);
        }
    }
}

#endif  // LIF_ASYNC

extern "C" void kernel_launch(void* const* d_in, const int* in_sizes, int n_in,
                              void* d_out, int out_size, void* d_ws, size_t ws_size,
                              hipStream_t stream) {
    const float* x = (const float*)d_in[0];
    float* out = (float*)d_out;

    const int grid = NTHREADS / BLOCK;   // 1024 blocks of 256 threads
    lif_scan_kernel<<<dim3(grid), dim3(BLOCK), 0, stream>>>(x, out);
}